// WsdmTripletLoss_39556648796742
// MI455X (gfx1250) — compile-verified
//
#include <hip/hip_runtime.h>
#include <hip/hip_bf16.h>

typedef __attribute__((ext_vector_type(2))) float v2f;
typedef __attribute__((ext_vector_type(8))) float v8f;

#define EPS_F 1e-8f
#define MARGIN_F 1.0f

// ---------------------------------------------------------------------------
// Kernel 1: exclusive scan of segment lengths -> offsets, per-row owner maps,
// and the total pair count. bs <= 1024, single workgroup.
// ---------------------------------------------------------------------------
__global__ __launch_bounds__(1024)
void scan_owner_kernel(const int* __restrict__ p_lens, const int* __restrict__ n_lens,
                       int bs, int* __restrict__ p_off, int* __restrict__ n_off,
                       int* __restrict__ owner_p, int* __restrict__ owner_n,
                       int* __restrict__ total_pairs) {
  __shared__ int sp[1024];
  __shared__ int sn[1024];
  const int t = threadIdx.x;
  const int pl = (t < bs) ? p_lens[t] : 0;
  const int nl = (t < bs) ? n_lens[t] : 0;
  sp[t] = pl;
  sn[t] = nl;
  __syncthreads();
  // Hillis-Steele inclusive scan
  for (int d = 1; d < 1024; d <<= 1) {
    const int vp = (t >= d) ? sp[t - d] : 0;
    const int vn = (t >= d) ? sn[t - d] : 0;
    __syncthreads();
    sp[t] += vp;
    sn[t] += vn;
    __syncthreads();
  }
  const int pexc = sp[t] - pl;   // exclusive prefix
  const int nexc = sn[t] - nl;
  if (t < bs) {
    p_off[t] = pexc;
    n_off[t] = nexc;
    for (int j = 0; j < pl; ++j) owner_p[pexc + j] = t;
    for (int k = 0; k < nl; ++k) owner_n[nexc + k] = t;
  }
  __syncthreads();
  sp[t] = pl * nl;
  __syncthreads();
  for (int d = 512; d > 0; d >>= 1) {
    if (t < d) sp[t] += sp[t + d];
    __syncthreads();
  }
  if (t == 0) total_pairs[0] = sp[0];
}

// ---------------------------------------------------------------------------
// Kernel 2: anchor L2 norms, one wave32 per row.
// ---------------------------------------------------------------------------
__global__ __launch_bounds__(256)
void anchor_norm_kernel(const float* __restrict__ anchor, float* __restrict__ a_norm,
                        int bs, int dim) {
  const int wave = (int)((blockIdx.x * blockDim.x + threadIdx.x) >> 5);
  const int lane = threadIdx.x & 31;
  if (wave >= bs) return;
  const float* __restrict__ row = anchor + (size_t)wave * dim;
  float s = 0.0f;
  for (int i = lane; i < dim; i += 32) {
    const float v = row[i];
    s += v * v;
  }
  for (int off = 16; off > 0; off >>= 1) s += __shfl_xor(s, off, 32);
  if (lane == 0) a_norm[wave] = sqrtf(s);
}

// ---------------------------------------------------------------------------
// Kernel 3: cosine distance per row via V_WMMA_F32_16X16X4_F32.
//
// Each wave owns a 16-row tile. Per K-chunk of 4:
//   A-matrix = (x ⊙ y) chunk, documented f32 16x4 layout:
//     M = lane&15, K = vgpr + 2*(lane>>4)  -> one float2 load per lane.
//   B-matrix = all ones (value identical in every lane, so the result is
//     independent of the B VGPR layout): D[m][n] = row-sum for every n.
// Two accumulators: dot(a,p) and dot(p,p). Diagonal-free extraction from the
// documented C layout: lanes 0-7 hold rows 0-7 (vgpr = lane), lanes 16-23
// hold rows 8-15 (vgpr = lane-16).
// ---------------------------------------------------------------------------
__device__ __forceinline__ float pick8(v8f v, int sel) {
  float r = v[0];
  if (sel == 1) r = v[1];
  if (sel == 2) r = v[2];
  if (sel == 3) r = v[3];
  if (sel == 4) r = v[4];
  if (sel == 5) r = v[5];
  if (sel == 6) r = v[6];
  if (sel == 7) r = v[7];
  return r;
}

__global__ __launch_bounds__(256)
void cosdist_wmma_kernel(const float* __restrict__ anchor,
                         const float* __restrict__ rows,
                         const int* __restrict__ owner,
                         const float* __restrict__ a_norm,
                         float* __restrict__ dist,
                         int nrows, int dim) {
  const int lane = threadIdx.x & 31;
  const int tile = (int)(blockIdx.x * (blockDim.x >> 5) + (threadIdx.x >> 5));
  const int base = tile * 16;
  if (base >= nrows) return;  // whole wave exits uniformly (EXEC stays all-1s)

  int r = base + (lane & 15);
  if (r > nrows - 1) r = nrows - 1;  // clamp tail rows (stores skipped below)
  const int own = owner[r];
  const int koff = (lane >> 4) << 1;  // 0 for lanes 0-15, 2 for lanes 16-31

  const float* __restrict__ ap = anchor + (size_t)own * dim + koff;
  const float* __restrict__ rp = rows + (size_t)r * dim + koff;

  const v2f ones = {1.0f, 1.0f};
  v8f acc_ap = {};
  v8f acc_pp = {};
#pragma unroll 8
  for (int k0 = 0; k0 < dim; k0 += 4) {
    const v2f a2 = *(const v2f*)(ap + k0);
    const v2f p2 = *(const v2f*)(rp + k0);
    const v2f zap = a2 * p2;  // elementwise a*p chunk
    const v2f zpp = p2 * p2;  // elementwise p*p chunk
    acc_ap = __builtin_amdgcn_wmma_f32_16x16x4_f32(
        false, zap, false, ones, (short)0, acc_ap, false, false);
    acc_pp = __builtin_amdgcn_wmma_f32_16x16x4_f32(
        false, zpp, false, ones, (short)0, acc_pp, false, false);
  }

  // Extract row sums: lane groups {0-7} -> rows 0-7, {16-23} -> rows 8-15.
  const int sel = lane & 7;
  const int m = sel + ((lane >> 4) << 3);
  const int out_r = base + m;
  if (((lane & 8) == 0) && out_r < nrows) {
    const float dot_ap = pick8(acc_ap, sel);
    const float dot_pp = pick8(acc_pp, sel);
    const float na = a_norm[owner[out_r]];
    const float np = sqrtf(dot_pp);
    const float c = dot_ap / fmaxf(na * np, EPS_F);
    dist[out_r] = 1.0f - c;
  }
}

// ---------------------------------------------------------------------------
// Kernel 4: per-batch hinge sum over p_i x n_i pairs, one wave per batch.
// ---------------------------------------------------------------------------
__global__ __launch_bounds__(256)
void pairloss_kernel(const float* __restrict__ d_ap, const float* __restrict__ d_an,
                     const int* __restrict__ p_off, const int* __restrict__ n_off,
                     const int* __restrict__ p_lens, const int* __restrict__ n_lens,
                     float* __restrict__ batch_loss, int bs) {
  const int wave = (int)((blockIdx.x * blockDim.x + threadIdx.x) >> 5);
  const int lane = threadIdx.x & 31;
  if (wave >= bs) return;
  const int pl = p_lens[wave], nl = n_lens[wave];
  const int po = p_off[wave], no = n_off[wave];
  const int tot = pl * nl;
  float s = 0.0f;
  for (int t = lane; t < tot; t += 32) {
    const int j = t / nl;
    const int k = t - j * nl;
    const float v = d_ap[po + j] - d_an[no + k] + MARGIN_F;
    s += fmaxf(v, 0.0f);
  }
  for (int off = 16; off > 0; off >>= 1) s += __shfl_xor(s, off, 32);
  if (lane == 0) batch_loss[wave] = s;
}

// ---------------------------------------------------------------------------
// Kernel 5: deterministic final reduce + mean.
// ---------------------------------------------------------------------------
__global__ __launch_bounds__(256)
void final_reduce_kernel(const float* __restrict__ batch_loss,
                         const int* __restrict__ total_pairs,
                         float* __restrict__ out, int bs) {
  __shared__ float sm[256];
  const int t = threadIdx.x;
  float s = 0.0f;
  for (int i = t; i < bs; i += 256) s += batch_loss[i];
  sm[t] = s;
  __syncthreads();
  for (int d = 128; d > 0; d >>= 1) {
    if (t < d) sm[t] += sm[t + d];
    __syncthreads();
  }
  if (t == 0) out[0] = sm[0] / (float)total_pairs[0];
}

extern "C" void kernel_launch(void* const* d_in, const int* in_sizes, int n_in,
                              void* d_out, int out_size, void* d_ws, size_t ws_size,
                              hipStream_t stream) {
  const float* anchor = (const float*)d_in[0];
  const float* positive = (const float*)d_in[1];
  const float* negative = (const float*)d_in[2];
  const int* p_lens = (const int*)d_in[3];
  const int* n_lens = (const int*)d_in[4];

  const int bs = in_sizes[3];
  const int dim = in_sizes[0] / bs;          // 768
  const int P = in_sizes[1] / dim;
  const int N = in_sizes[2] / dim;

  // Workspace carve-up (256B aligned slices).
  size_t cur = 0;
  auto walloc = [&](size_t nbytes) -> char* {
    char* p = (char*)d_ws + cur;
    cur += (nbytes + 255) & ~(size_t)255;
    return p;
  };
  int* p_off = (int*)walloc((size_t)bs * 4);
  int* n_off = (int*)walloc((size_t)bs * 4);
  int* owner_p = (int*)walloc((size_t)P * 4);
  int* owner_n = (int*)walloc((size_t)N * 4);
  float* a_norm = (float*)walloc((size_t)bs * 4);
  float* dAP = (float*)walloc((size_t)P * 4);
  float* dAN = (float*)walloc((size_t)N * 4);
  float* batch_loss = (float*)walloc((size_t)bs * 4);
  int* total_pairs = (int*)walloc(4);
  (void)ws_size; (void)n_in; (void)out_size;

  scan_owner_kernel<<<1, 1024, 0, stream>>>(p_lens, n_lens, bs, p_off, n_off,
                                            owner_p, owner_n, total_pairs);
  anchor_norm_kernel<<<(bs + 7) / 8, 256, 0, stream>>>(anchor, a_norm, bs, dim);

  const int tilesP = (P + 15) / 16;
  const int tilesN = (N + 15) / 16;
  cosdist_wmma_kernel<<<(tilesP + 7) / 8, 256, 0, stream>>>(
      anchor, positive, owner_p, a_norm, dAP, P, dim);
  cosdist_wmma_kernel<<<(tilesN + 7) / 8, 256, 0, stream>>>(
      anchor, negative, owner_n, a_norm, dAN, N, dim);

  pairloss_kernel<<<(bs + 7) / 8, 256, 0, stream>>>(dAP, dAN, p_off, n_off,
                                                    p_lens, n_lens, batch_loss, bs);
  final_reduce_kernel<<<1, 256, 0, stream>>>(batch_loss, total_pairs,
                                             (float*)d_out, bs);
}